// MemoryGNN_44203803410970
// MI455X (gfx1250) — compile-verified
//
#include <hip/hip_runtime.h>
#include <hip/hip_bf16.h>
#include <math.h>

// ---------------------------------------------------------------------------
// Problem constants (match reference)
// ---------------------------------------------------------------------------
#define NN     50000LL      // nodes  (50000 = 3125 * 16, exact WMMA M tiles)
#define EE     1000000LL    // edges
#define EP     (EE + NN)    // edges + self loops
#define INDIM  256
#define HID    64
#define HEADS  4
#define OUTDIM 128
#define HC1    (HEADS * HID)   // 256
#define NEG_SLOPE 0.2f

typedef float v2f __attribute__((ext_vector_type(2)));
typedef float v8f __attribute__((ext_vector_type(8)));

// ---------------------------------------------------------------------------
// Generic fp32 WMMA GEMM:  C[M,Nc] = act(A[M,K] @ B[K,Nc] + bias)
// One wave computes one 16x16 output tile via V_WMMA_F32_16X16X4_F32.
// Requires: M % 16 == 0, Nc % 16 == 0, K % 4 == 0  (true for all our GEMMs).
// blockDim = 128 (4 waves); grid = (ceil(M/64), Nc/16).
// act: 0 = none, 1 = relu
// ---------------------------------------------------------------------------
__global__ __launch_bounds__(128)
void wmma_gemm_f32(const float* __restrict__ A, const float* __restrict__ B,
                   const float* __restrict__ bias, float* __restrict__ C,
                   int M, int Nc, int K, int act) {
  const int lane  = threadIdx.x & 31;
  const int wave  = threadIdx.x >> 5;
  const int tileM = blockIdx.x * 4 + wave;
  if (tileM * 16 >= M) return;                 // wave-uniform: EXEC stays all-1
  const int m0   = tileM * 16;
  const int n0   = blockIdx.y * 16;
  const int half = lane >> 4;                  // 0: lanes 0-15, 1: lanes 16-31
  const int l    = lane & 15;

  const float* Arow = A + (size_t)(m0 + l) * K;   // lane l holds row m0+l
  const float* Bcol = B + n0 + l;                 // lane l holds col n0+l

  v8f acc = {};
  for (int k = 0; k < K; k += 4) {
    const int ka = k + 2 * half;               // K{0,1} lower half, K{2,3} upper
    v2f a = *(const v2f*)(Arow + ka);          // 8B aligned (K % 4 == 0)
    v2f b;
    b.x = Bcol[(size_t)ka * Nc];
    b.y = Bcol[(size_t)(ka + 1) * Nc];
    acc = __builtin_amdgcn_wmma_f32_16x16x4_f32(
        /*neg_a=*/false, a, /*neg_b=*/false, b,
        /*c_mod=*/(short)0, acc, /*reuse_a=*/false, /*reuse_b=*/false);
  }

  const float bv = bias ? bias[n0 + l] : 0.0f;
  #pragma unroll
  for (int r = 0; r < 8; ++r) {                // VGPR r: row r (lower) / r+8 (upper)
    float v = acc[r] + bv;
    if (act == 1) v = fmaxf(v, 0.0f);
    C[(size_t)(m0 + r + 8 * half) * Nc + n0 + l] = v;
  }
}

// ---------------------------------------------------------------------------
// Monotone float<->uint key for atomicMax-based segment max
// ---------------------------------------------------------------------------
__device__ __forceinline__ unsigned fkey(float f) {
  unsigned b = __float_as_uint(f);
  return (b & 0x80000000u) ? ~b : (b | 0x80000000u);
}
__device__ __forceinline__ float keyf(unsigned k) {
  unsigned b = (k & 0x80000000u) ? (k & 0x7FFFFFFFu) : ~k;
  return __uint_as_float(b);
}

// ---------------------------------------------------------------------------
// Layer-1 attention scores: s_src/s_dst [N,4] from h1 [N,256]
// ---------------------------------------------------------------------------
__global__ void scores1_kernel(const float* __restrict__ h1,
                               const float* __restrict__ a1s,
                               const float* __restrict__ a1d,
                               float* __restrict__ ssrc, float* __restrict__ sdst) {
  long long t = blockIdx.x * (long long)blockDim.x + threadIdx.x;  // node*4+head
  if (t >= NN * HEADS) return;
  const long long n = t >> 2;
  const int h = (int)(t & 3);
  const float* row = h1 + n * HC1 + h * HID;
  const float* as = a1s + h * HID;
  const float* ad = a1d + h * HID;
  float s0 = 0.f, s1 = 0.f;
  #pragma unroll 8
  for (int c = 0; c < HID; ++c) { float v = row[c]; s0 += v * as[c]; s1 += v * ad[c]; }
  ssrc[t] = s0; sdst[t] = s1;
}

__global__ void scores2_kernel(const float* __restrict__ h2,
                               const float* __restrict__ a2s,
                               const float* __restrict__ a2d,
                               float* __restrict__ ssrc, float* __restrict__ sdst) {
  long long n = blockIdx.x * (long long)blockDim.x + threadIdx.x;
  if (n >= NN) return;
  const float* row = h2 + n * OUTDIM;
  float s0 = 0.f, s1 = 0.f;
  #pragma unroll 8
  for (int c = 0; c < OUTDIM; ++c) { float v = row[c]; s0 += v * a2s[c]; s1 += v * a2d[c]; }
  ssrc[n] = s0; sdst[n] = s1;
}

// ---------------------------------------------------------------------------
// Edge pass A: leaky-relu score + segment max (atomicMax on monotone key)
// ---------------------------------------------------------------------------
__global__ void edge_max1(const long long* __restrict__ ei,
                          const float* __restrict__ ssrc,
                          const float* __restrict__ sdst,
                          float* __restrict__ e1, unsigned* __restrict__ m1) {
  long long t = blockIdx.x * (long long)blockDim.x + threadIdx.x;  // edge*4+head
  if (t >= EP * HEADS) return;
  const long long e = t >> 2;
  const int h = (int)(t & 3);
  long long s, d;
  if (e < EE) { s = ei[e]; d = ei[EE + e]; } else { s = d = e - EE; }
  float v = ssrc[s * HEADS + h] + sdst[d * HEADS + h];
  v = (v > 0.f) ? v : NEG_SLOPE * v;
  e1[t] = v;
  atomicMax(&m1[d * HEADS + h], fkey(v));
}

__global__ void edge_max2(const long long* __restrict__ ei,
                          const float* __restrict__ ssrc,
                          const float* __restrict__ sdst,
                          float* __restrict__ e2, unsigned* __restrict__ m2) {
  long long e = blockIdx.x * (long long)blockDim.x + threadIdx.x;
  if (e >= EP) return;
  long long s, d;
  if (e < EE) { s = ei[e]; d = ei[EE + e]; } else { s = d = e - EE; }
  float v = ssrc[s] + sdst[d];
  v = (v > 0.f) ? v : NEG_SLOPE * v;
  e2[e] = v;
  atomicMax(&m2[d], fkey(v));
}

// ---------------------------------------------------------------------------
// Edge pass B: exp(e - max) + segment sum
// ---------------------------------------------------------------------------
__global__ void edge_exp1(const long long* __restrict__ ei,
                          const unsigned* __restrict__ m1,
                          float* __restrict__ e1, float* __restrict__ denom1) {
  long long t = blockIdx.x * (long long)blockDim.x + threadIdx.x;
  if (t >= EP * HEADS) return;
  const long long e = t >> 2;
  const int h = (int)(t & 3);
  long long d;
  if (e < EE) { d = ei[EE + e]; } else { d = e - EE; }
  float x = expf(e1[t] - keyf(m1[d * HEADS + h]));
  e1[t] = x;
  atomicAdd(&denom1[d * HEADS + h], x);
}

__global__ void edge_exp2(const long long* __restrict__ ei,
                          const unsigned* __restrict__ m2,
                          float* __restrict__ e2, float* __restrict__ denom2) {
  long long e = blockIdx.x * (long long)blockDim.x + threadIdx.x;
  if (e >= EP) return;
  long long d;
  if (e < EE) { d = ei[EE + e]; } else { d = e - EE; }
  float x = expf(e2[e] - keyf(m2[d]));
  e2[e] = x;
  atomicAdd(&denom2[d], x);
}

// ---------------------------------------------------------------------------
// Edge pass C: weighted scatter-add of messages (block per edge)
// Working set (h + out) fits in the 192MB L2 -> atomics mostly L2-resident.
// ---------------------------------------------------------------------------
__global__ __launch_bounds__(HC1)
void agg1(const long long* __restrict__ ei, const float* __restrict__ h1,
          const float* __restrict__ e1, const float* __restrict__ denom1,
          float* __restrict__ out1) {
  const long long e = blockIdx.x;
  const int c = threadIdx.x;                 // 0..255
  long long s, d;
  if (e < EE) { s = ei[e]; d = ei[EE + e]; } else { s = d = e - EE; }
  const int h = c >> 6;
  const float alpha = e1[e * HEADS + h] / (denom1[d * HEADS + h] + 1e-16f);
  atomicAdd(&out1[d * HC1 + c], h1[s * HC1 + c] * alpha);
}

__global__ __launch_bounds__(OUTDIM)
void agg2(const long long* __restrict__ ei, const float* __restrict__ h2,
          const float* __restrict__ e2, const float* __restrict__ denom2,
          float* __restrict__ out2) {
  const long long e = blockIdx.x;
  const int c = threadIdx.x;                 // 0..127
  long long s, d;
  if (e < EE) { s = ei[e]; d = ei[EE + e]; } else { s = d = e - EE; }
  const float alpha = e2[e] / (denom2[d] + 1e-16f);
  atomicAdd(&out2[d * OUTDIM + c], h2[s * OUTDIM + c] * alpha);
}

// ---------------------------------------------------------------------------
// Epilogues: bias + ELU (layer 1), bias only (layer 2)
// ---------------------------------------------------------------------------
__global__ void finish1_kernel(float* __restrict__ out1, const float* __restrict__ b1) {
  long long t = blockIdx.x * (long long)blockDim.x + threadIdx.x;
  if (t >= NN * HC1) return;
  float v = out1[t] + b1[(int)(t & (HC1 - 1))];
  out1[t] = (v > 0.f) ? v : (expf(v) - 1.0f);       // ELU(alpha=1)
}

__global__ void finish2_kernel(float* __restrict__ out2, const float* __restrict__ b2) {
  long long t = blockIdx.x * (long long)blockDim.x + threadIdx.x;
  if (t >= NN * OUTDIM) return;
  out2[t] = out2[t] + b2[(int)(t & (OUTDIM - 1))];  // heads=1 mean == identity
}

// ---------------------------------------------------------------------------
// Row L2-normalize (wave per row of 128)
// ---------------------------------------------------------------------------
__global__ __launch_bounds__(256)
void normalize_kernel(const float* __restrict__ h4, float* __restrict__ out) {
  const long long row = (blockIdx.x * (long long)blockDim.x + threadIdx.x) >> 5;
  const int lane = threadIdx.x & 31;
  if (row >= NN) return;
  const float* r = h4 + row * OUTDIM;
  float v0 = r[lane], v1 = r[lane + 32], v2 = r[lane + 64], v3 = r[lane + 96];
  float s = v0 * v0 + v1 * v1 + v2 * v2 + v3 * v3;
  #pragma unroll
  for (int off = 16; off > 0; off >>= 1) s += __shfl_xor(s, off, 32);
  const float inv = 1.0f / fmaxf(sqrtf(s), 1e-12f);
  float* o = out + row * OUTDIM;
  o[lane] = v0 * inv; o[lane + 32] = v1 * inv; o[lane + 64] = v2 * inv; o[lane + 96] = v3 * inv;
}

// ---------------------------------------------------------------------------
// Launch
// ---------------------------------------------------------------------------
extern "C" void kernel_launch(void* const* d_in, const int* in_sizes, int n_in,
                              void* d_out, int out_size, void* d_ws, size_t ws_size,
                              hipStream_t stream) {
  const float*     x   = (const float*)d_in[0];
  const long long* ei  = (const long long*)d_in[1];   // int64 [2,E]
  const float*     W1  = (const float*)d_in[2];
  const float*     a1s = (const float*)d_in[3];
  const float*     a1d = (const float*)d_in[4];
  const float*     b1  = (const float*)d_in[5];
  const float*     W2  = (const float*)d_in[6];
  const float*     a2s = (const float*)d_in[7];
  const float*     a2d = (const float*)d_in[8];
  const float*     b2  = (const float*)d_in[9];
  const float*     Wm1 = (const float*)d_in[10];
  const float*     bm1 = (const float*)d_in[11];
  const float*     Wm2 = (const float*)d_in[12];
  const float*     bm2 = (const float*)d_in[13];

  // ---- workspace layout (floats) ----
  float* ws = (float*)d_ws;
  size_t off = 0;
  // zero-initialized contiguous block: m1, denom1, out1, m2, denom2, out2
  unsigned* m1     = (unsigned*)(ws + off);  off += NN * HEADS;
  float*    denom1 = ws + off;               off += NN * HEADS;
  float*    out1   = ws + off;               off += NN * HC1;       // -> h1act in place
  unsigned* m2     = (unsigned*)(ws + off);  off += NN;
  float*    denom2 = ws + off;               off += NN;
  float*    out2   = ws + off;               off += NN * OUTDIM;
  const size_t zero_elems = off;
  // non-zeroed scratch
  float* h1    = ws + off;  off += NN * HC1;
  float* ssrc1 = ws + off;  off += NN * HEADS;
  float* sdst1 = ws + off;  off += NN * HEADS;
  float* e1    = ws + off;  off += EP * HEADS;
  float* h2    = ws + off;  off += NN * OUTDIM;
  float* ssrc2 = ws + off;  off += NN;
  float* sdst2 = ws + off;  off += NN;
  float* e2    = ws + off;  off += EP;
  float* h3    = ws + off;  off += NN * HID;
  float* h4    = ws + off;  off += NN * OUTDIM;
  (void)ws_size; (void)in_sizes; (void)n_in; (void)out_size;

  hipMemsetAsync(d_ws, 0, zero_elems * sizeof(float), stream);

  const int MT = (int)((NN / 16 + 3) / 4);   // 782 blocks of 4 waves

  // Layer 1: h1 = x @ W1   [50000,256] = [50000,256]@[256,256]
  wmma_gemm_f32<<<dim3(MT, HC1 / 16), 128, 0, stream>>>(x, W1, nullptr, h1,
                                                        (int)NN, HC1, INDIM, 0);
  scores1_kernel<<<(unsigned)((NN * HEADS + 255) / 256), 256, 0, stream>>>(h1, a1s, a1d, ssrc1, sdst1);
  edge_max1<<<(unsigned)((EP * HEADS + 255) / 256), 256, 0, stream>>>(ei, ssrc1, sdst1, e1, m1);
  edge_exp1<<<(unsigned)((EP * HEADS + 255) / 256), 256, 0, stream>>>(ei, m1, e1, denom1);
  agg1<<<(unsigned)EP, HC1, 0, stream>>>(ei, h1, e1, denom1, out1);
  finish1_kernel<<<(unsigned)((NN * HC1 + 255) / 256), 256, 0, stream>>>(out1, b1);

  // Layer 2: h2 = elu_out @ W2   [50000,128] = [50000,256]@[256,128]
  wmma_gemm_f32<<<dim3(MT, OUTDIM / 16), 128, 0, stream>>>(out1, W2, nullptr, h2,
                                                           (int)NN, OUTDIM, HC1, 0);
  scores2_kernel<<<(unsigned)((NN + 255) / 256), 256, 0, stream>>>(h2, a2s, a2d, ssrc2, sdst2);
  edge_max2<<<(unsigned)((EP + 255) / 256), 256, 0, stream>>>(ei, ssrc2, sdst2, e2, m2);
  edge_exp2<<<(unsigned)((EP + 255) / 256), 256, 0, stream>>>(ei, m2, e2, denom2);
  agg2<<<(unsigned)EP, OUTDIM, 0, stream>>>(ei, h2, e2, denom2, out2);
  finish2_kernel<<<(unsigned)((NN * OUTDIM + 255) / 256), 256, 0, stream>>>(out2, b2);

  // MLP: h3 = relu(out2 @ Wm1 + bm1); h4 = h3 @ Wm2 + bm2
  wmma_gemm_f32<<<dim3(MT, HID / 16), 128, 0, stream>>>(out2, Wm1, bm1, h3,
                                                        (int)NN, HID, OUTDIM, 1);
  wmma_gemm_f32<<<dim3(MT, OUTDIM / 16), 128, 0, stream>>>(h3, Wm2, bm2, h4,
                                                           (int)NN, OUTDIM, HID, 0);

  // Row-normalize -> d_out
  normalize_kernel<<<(unsigned)((NN * 32 + 255) / 256), 256, 0, stream>>>(h4, (float*)d_out);
}